// EnhancedStockPredictor_29978871726239
// MI455X (gfx1250) — compile-verified
//
#include <hip/hip_runtime.h>
#include <hip/hip_bf16.h>
#include <math.h>
#include <stdint.h>

// ---- model constants -------------------------------------------------------
#define BB   32
#define TT   512
#define FF   32
#define HH   512
#define EE   8
#define EDD  128
#define MAXKK 50
#define ALPHA_ 0.3f
#define BETA_  0.4f
#define ETA_   0.3f

#define KT      32      // K-chunk per pipeline stage
#define ASTRIDE 33      // padded LDS row stride for K-major tiles (bank-conflict-free)

typedef float v2f __attribute__((ext_vector_type(2)));
typedef float v8f __attribute__((ext_vector_type(8)));
typedef unsigned int u32x4 __attribute__((ext_vector_type(4)));
typedef int i32x4 __attribute__((ext_vector_type(4)));
typedef int i32x8 __attribute__((ext_vector_type(8)));

__device__ __forceinline__ float sigm(float x) { return 1.0f / (1.0f + expf(-x)); }

// ---------------------------------------------------------------------------
// Tensor Data Mover: async 2-D tile load global->LDS (CDNA5 ISA).
// D# group0: count=1 | lds_addr | global_addr[56:0] | type=2.
// D# group1: data_size=4B, tile (cols x rows), tensor dims == tile (no OOB),
//            tensor_dim0_stride = ld;  optional LDS pad: +1 dword per 32 dwords.
// Tracked by TENSORcnt; caller waits with s_wait_tensorcnt.
// ---------------------------------------------------------------------------
template <bool PAD>
__device__ __forceinline__ void tdm_load_2d(uint32_t lds_off, const float* gptr,
                                            uint32_t rows, uint32_t cols,
                                            uint64_t stride_elems)
{
  const uint64_t ga = (uint64_t)(uintptr_t)gptr;
  u32x4 g0;
  g0[0] = 1u;                                   // count=1, user-mode, no gather
  g0[1] = lds_off;                              // lds_addr (bytes)
  g0[2] = (uint32_t)ga;                         // global_addr[31:0]
  g0[3] = (uint32_t)((ga >> 32) & 0x1FFFFFFu) | (2u << 30); // addr[56:32] | type=2

  uint32_t w0 = (2u << 16);                     // data_size = 4 bytes
  if (PAD) w0 |= (1u << 20) | (4u << 22);       // pad_enable, interval=32 dw, amount=1 dw
  i32x8 g1;
  g1[0] = (int)w0;
  g1[1] = (int)((cols & 0xFFFFu) << 16);                          // tensor_dim0 lo
  g1[2] = (int)(((cols >> 16) & 0xFFFFu) | ((rows & 0xFFFFu) << 16)); // dim0 hi | dim1 lo
  g1[3] = (int)(((rows >> 16) & 0xFFFFu) | ((cols & 0xFFFFu) << 16)); // dim1 hi | tile_dim0
  g1[4] = (int)(rows & 0xFFFFu);                                  // tile_dim1 (tile_dim2=0)
  g1[5] = (int)(uint32_t)(stride_elems & 0xFFFFFFFFu);            // dim0_stride lo
  g1[6] = (int)(uint32_t)((stride_elems >> 32) & 0xFFFFu);        // dim0_stride hi
  g1[7] = 0;

  i32x4 z4 = {0, 0, 0, 0};
#if __clang_major__ >= 23
  i32x8 z8 = {0, 0, 0, 0, 0, 0, 0, 0};
  __builtin_amdgcn_tensor_load_to_lds(g0, g1, z4, z4, z8, 0);
#else
  __builtin_amdgcn_tensor_load_to_lds(g0, g1, z4, z4, 0);
#endif
}

__device__ __forceinline__ uint32_t lds_off_of(const void* p)
{
  // flat LDS aperture: addr[31:0] is the LDS byte offset
  return (uint32_t)(uintptr_t)p;
}

// ---------------------------------------------------------------------------
// fp32 WMMA GEMM:  C = act( ([C +] A(MxK) * B(KxN or NxK^T) + bias) * rscale )
// Block = 128 threads (4 waves) -> 16(M) x 64(N) tile.  Grid = (N/64, M/16).
// K pipelined in KT=32 chunks, double-buffered in LDS, staged by the TDM
// (wave 0 issues tensor_load_to_lds; s_wait_tensorcnt retires stage s while
// stage s+1 is in flight).  8 x V_WMMA_F32_16X16X4_F32 per stage per wave.
// ---------------------------------------------------------------------------
template <int BT, int ACCUM, int ACT>
__global__ __launch_bounds__(128) void gemm_wmma_f32(
    const float* __restrict__ A, int lda,
    const float* __restrict__ B, int ldb,
    float* __restrict__ C, int ldc,
    const float* __restrict__ bias,
    const float* __restrict__ rscale,
    int K)
{
  __shared__ float As[2][16 * ASTRIDE];                       // K-major, padded
  __shared__ float Bs[2][64 * ASTRIDE];                       // max(BT:64x33, !BT:32x64)

  const int lane = threadIdx.x & 31;
  const int wave = threadIdx.x >> 5;
  const int m0   = blockIdx.y * 16;
  const int n0b  = blockIdx.x * 64;
  const int half = lane >> 4;            // K sub-pair selector per WMMA layout
  const int l16  = lane & 15;
  const int coln = wave * 16 + l16;      // column inside the 64-wide block tile
  const bool lead = (threadIdx.x < 32);  // wave 0 drives the TDM
  const int S = K / KT;

  v8f acc = {};
  if (ACCUM) {
#pragma unroll
    for (int r = 0; r < 8; ++r)
      acc[r] = C[(m0 + r + half * 8) * ldc + n0b + coln - wave * 16 + wave * 16]; // row,col
  }
  // (the expression above is just C[(m0+r+half*8)*ldc + (n0b+coln)])

  if (lead) {
    tdm_load_2d<true>(lds_off_of(&As[0][0]), A + (size_t)m0 * lda, 16, KT, (uint64_t)lda);
    if (BT) tdm_load_2d<true >(lds_off_of(&Bs[0][0]), B + (size_t)n0b * ldb, 64, KT, (uint64_t)ldb);
    else    tdm_load_2d<false>(lds_off_of(&Bs[0][0]), B + n0b,               KT, 64, (uint64_t)ldb);
  }

  for (int s = 0; s < S; ++s) {
    if (lead) {
      const int kb = (s + 1) * KT;
      if (s + 1 < S) {
        const int nb = (s + 1) & 1;
        tdm_load_2d<true>(lds_off_of(&As[nb][0]), A + (size_t)m0 * lda + kb, 16, KT, (uint64_t)lda);
        if (BT) tdm_load_2d<true >(lds_off_of(&Bs[nb][0]), B + (size_t)n0b * ldb + kb, 64, KT, (uint64_t)ldb);
        else    tdm_load_2d<false>(lds_off_of(&Bs[nb][0]), B + (size_t)kb * ldb + n0b, KT, 64, (uint64_t)ldb);
        __builtin_amdgcn_s_wait_tensorcnt(2);   // stage s resident; s+1 still in flight
      } else {
        __builtin_amdgcn_s_wait_tensorcnt(0);
      }
    }
    __syncthreads();

    const float* a_s = &As[s & 1][0];
    const float* b_s = &Bs[s & 1][0];
#pragma unroll
    for (int k0 = 0; k0 < KT; k0 += 4) {
      const int ka = k0 + half * 2;
      v2f a, b;
      a[0] = a_s[l16 * ASTRIDE + ka];
      a[1] = a_s[l16 * ASTRIDE + ka + 1];
      if (BT) {
        b[0] = b_s[coln * ASTRIDE + ka];
        b[1] = b_s[coln * ASTRIDE + ka + 1];
      } else {
        b[0] = b_s[ka * 64 + coln];
        b[1] = b_s[(ka + 1) * 64 + coln];
      }
      acc = __builtin_amdgcn_wmma_f32_16x16x4_f32(
          false, a, false, b, (short)0, acc, false, false);
    }
    __syncthreads();
  }

#pragma unroll
  for (int r = 0; r < 8; ++r) {
    const int row = m0 + r + half * 8;
    const int col = n0b + coln;
    float v = acc[r];
    if (bias)   v += bias[col];
    if (rscale) v *= rscale[row];
    if (ACT == 1) v = fmaxf(v, 0.0f);
    C[row * ldc + col] = v;
  }
}

// ---- row-wise softmax of W_mcao with zeroed diagonal ----------------------
__global__ void softmax_w_kernel(const float* __restrict__ W, float* __restrict__ wsoft)
{
  int i = threadIdx.x;            // 32 rows, 32 threads
  float mx = -1e30f;
  for (int j = 0; j < FF; ++j) mx = fmaxf(mx, W[i * FF + j]);
  float e[FF];
  float s = 0.0f;
  for (int j = 0; j < FF; ++j) { e[j] = expf(W[i * FF + j] - mx); s += e[j]; }
  float inv = 1.0f / s;
  for (int j = 0; j < FF; ++j) wsoft[i * FF + j] = (i == j) ? 0.0f : e[j] * inv;
}

// ---- mem / coup / glob / tmdo + write concat X1 (wave32 shuffle) ----------
__global__ void tmdo_kernel(const float* __restrict__ x,
                            const float* __restrict__ coeffs,
                            const float* __restrict__ wsoft,
                            const float* __restrict__ Wg,
                            const float* __restrict__ bg,
                            const float* __restrict__ kappa_p,
                            float* __restrict__ tmdo_out,
                            float* __restrict__ mem_ws,
                            float* __restrict__ X1)
{
  const int bt = blockIdx.x;
  const int b = bt / TT, t = bt % TT;
  const int f = threadIdx.x;      // 0..31
  const float xf = x[(b * TT + t) * FF + f];

  float mem = 0.0f;
  const int kmax = (t < MAXKK) ? t : MAXKK;
  for (int k = 0; k <= kmax; ++k)
    mem += coeffs[k] * x[(b * TT + (t - k)) * FF + f];

  float coup = 0.0f;
  const float axf = fabsf(xf);
  for (int j = 0; j < FF; ++j) {
    float xj = __shfl(xf, j, 32);
    float v  = tanhf(xf - xj) * sigm(axf + fabsf(xj));
    coup += v * wsoft[f * FF + j];
  }

  float gs = bg[f];
  for (int k = 0; k < FF; ++k) {
    float xk = __shfl(xf, k, 32);
    gs += xk * Wg[k * FF + f];
  }
  const float r    = expf(-kappa_p[0]);
  const float ksum = (1.0f - powf(r, (float)(t + 1))) / (1.0f - r);

  const float out = ALPHA_ * mem + BETA_ * coup + ETA_ * (gs * ksum);
  const int base = (b * TT + t);
  tmdo_out[base * FF + f] = out;
  mem_ws[base * FF + f]   = mem;
  X1[base * 3 * FF + f]          = xf;
  X1[base * 3 * FF + FF + f]     = out;
  X1[base * 3 * FF + 2 * FF + f] = mem;
}

// ---- pack Wk|Wv and bk|bv so K/V come from one GEMM -----------------------
__global__ void build_wkv_kernel(const float* __restrict__ Wk, const float* __restrict__ bk,
                                 const float* __restrict__ Wv, const float* __restrict__ bv,
                                 float* __restrict__ Wkv, float* __restrict__ bkv)
{
  int i = blockIdx.x * blockDim.x + threadIdx.x;
  if (i < HH * 2 * HH) {
    int k = i >> 10, n = i & 1023;
    Wkv[i] = (n < HH) ? Wk[k * HH + n] : Wv[k * HH + (n - HH)];
  }
  if (i < 2 * HH) bkv[i] = (i < HH) ? bk[i] : bv[i - HH];
}

__global__ void zero_kernel(float* __restrict__ p, int n)
{
  int i = blockIdx.x * blockDim.x + threadIdx.x;
  if (i < n) p[i] = 0.0f;
}

// ---- per-step embedding gather + decay row-scales -------------------------
__global__ void gather_emb_kernel(const int* __restrict__ events,
                                  const float* __restrict__ td,
                                  const float* __restrict__ E_emb,
                                  int t, float* __restrict__ G, float* __restrict__ rs)
{
  int m = blockIdx.x;             // b*E + e  (256 rows)
  int b = m >> 3, e = m & 7;
  int idx = events[(b * TT + t) * EE + e];
  const float* src = E_emb + idx * EDD;
  for (int d = threadIdx.x; d < EDD; d += blockDim.x) G[m * EDD + d] = src[d];
  if (threadIdx.x == 0) rs[m] = expf(-0.1f * td[(b * TT + t) * EE + e]);
}

// ---- 4-head attention over E=8 events -------------------------------------
__global__ void attn_kernel(const float* __restrict__ q,
                            const float* __restrict__ kkvv,
                            float* __restrict__ evr)
{
  const int b = blockIdx.x >> 2, hd = blockIdx.x & 3, d = threadIdx.x; // 128 threads
  __shared__ float red[128];
  __shared__ float sc[EE];
  const float qv = q[b * HH + hd * 128 + d];
  for (int e = 0; e < EE; ++e) {
    red[d] = qv * kkvv[(b * EE + e) * 1024 + hd * 128 + d];
    __syncthreads();
    for (int s = 64; s > 0; s >>= 1) { if (d < s) red[d] += red[d + s]; __syncthreads(); }
    if (d == 0) sc[e] = red[0] * 0.0883883476f;  // dh^-0.5, dh=128
    __syncthreads();
  }
  if (d == 0) {
    float mx = sc[0];
    for (int e = 1; e < EE; ++e) mx = fmaxf(mx, sc[e]);
    float s = 0.0f;
    for (int e = 0; e < EE; ++e) { sc[e] = expf(sc[e] - mx); s += sc[e]; }
    float inv = 1.0f / s;
    for (int e = 0; e < EE; ++e) sc[e] *= inv;
  }
  __syncthreads();
  float acc = 0.0f;
  for (int e = 0; e < EE; ++e)
    acc += sc[e] * kkvv[(b * EE + e) * 1024 + HH + hd * 128 + d];
  evr[b * HH + hd * 128 + d] = acc;
}

// ---- LSTM cell elementwise -------------------------------------------------
__global__ void lstm_cell_kernel(const float* __restrict__ gates,
                                 float* __restrict__ c, float* __restrict__ h2a)
{
  int i = blockIdx.x * blockDim.x + threadIdx.x;  // 32*512
  int b = i >> 9, n = i & (HH - 1);
  const float* g = gates + b * 4 * HH;
  float ig = sigm(g[n]);
  float fg = sigm(g[HH + n]);
  float gg = tanhf(g[2 * HH + n]);
  float og = sigm(g[3 * HH + n]);
  float c2 = fg * c[i] + ig * gg;
  c[i] = c2;
  h2a[i] = og * tanhf(c2);
}

// ---- gated mix: out = a*sig(glin) + alt*(1-sig(glin)) ---------------------
__global__ void gate_mix_kernel(const float* __restrict__ a,
                                const float* __restrict__ alt,
                                const float* __restrict__ glin,
                                float* __restrict__ out)
{
  int i = blockIdx.x * blockDim.x + threadIdx.x;
  float g = sigm(glin[i]);
  out[i] = a[i] * g + alt[i] * (1.0f - g);
}

// ---- final prediction dot (N=1) -------------------------------------------
__global__ void pred_kernel(const float* __restrict__ ph,
                            const float* __restrict__ Wp2,
                            const float* __restrict__ bp2,
                            float* __restrict__ preds, int t)
{
  int b = blockIdx.x, tid = threadIdx.x;          // 256 threads
  __shared__ float red[256];
  red[tid] = ph[b * (HH / 2) + tid] * Wp2[tid];
  __syncthreads();
  for (int s = 128; s > 0; s >>= 1) { if (tid < s) red[tid] += red[tid + s]; __syncthreads(); }
  if (tid == 0) preds[b * TT + t] = red[0] + bp2[0];
}

// ---------------------------------------------------------------------------
static void gemm(hipStream_t stream,
                 const float* A, int lda, const float* B, int ldb, int btrans,
                 float* C, int ldc, const float* bias, const float* rscale,
                 int M, int N, int K, int accum, int act)
{
  dim3 grid(N / 64, M / 16), blk(128);
#define GL(BT, AC, ACTV) \
  gemm_wmma_f32<BT, AC, ACTV><<<grid, blk, 0, stream>>>(A, lda, B, ldb, C, ldc, bias, rscale, K)
  if (!btrans && !accum && act == 0)      GL(0, 0, 0);
  else if (!btrans && !accum && act == 1) GL(0, 0, 1);
  else if (!btrans && accum)              GL(0, 1, 0);
  else if (btrans && !accum)              GL(1, 0, 0);
  else                                    GL(1, 1, 0);
#undef GL
}

extern "C" void kernel_launch(void* const* d_in, const int* in_sizes, int n_in,
                              void* d_out, int out_size, void* d_ws, size_t ws_size,
                              hipStream_t stream)
{
  // ---- inputs (setup_inputs dict order) ----
  const float* x       = (const float*)d_in[0];
  const int*   events  = (const int*)  d_in[1];
  const float* td      = (const float*)d_in[2];
  const float* coeffs  = (const float*)d_in[3];
  const float* W_mcao  = (const float*)d_in[4];
  const float* Wg      = (const float*)d_in[5];
  const float* bg      = (const float*)d_in[6];
  const float* kappa   = (const float*)d_in[7];
  const float* Wf1     = (const float*)d_in[8];
  const float* bf1     = (const float*)d_in[9];
  const float* Wf2     = (const float*)d_in[10];
  const float* bf2     = (const float*)d_in[11];
  const float* E_emb   = (const float*)d_in[12];
  const float* We1     = (const float*)d_in[13];
  const float* be1     = (const float*)d_in[14];
  const float* We2     = (const float*)d_in[15];
  const float* be2     = (const float*)d_in[16];
  const float* Wq      = (const float*)d_in[17];
  const float* bq      = (const float*)d_in[18];
  const float* Wk      = (const float*)d_in[19];
  const float* bk      = (const float*)d_in[20];
  const float* Wv      = (const float*)d_in[21];
  const float* bv      = (const float*)d_in[22];
  const float* Wo      = (const float*)d_in[23];
  const float* bo      = (const float*)d_in[24];
  const float* W_ih    = (const float*)d_in[25];
  const float* b_ih    = (const float*)d_in[26];
  const float* W_hh    = (const float*)d_in[27];
  const float* b_hh    = (const float*)d_in[28];
  const float* Wip     = (const float*)d_in[29];
  const float* bip     = (const float*)d_in[30];
  const float* Wig     = (const float*)d_in[31];
  const float* big     = (const float*)d_in[32];
  const float* Weg     = (const float*)d_in[33];
  const float* beg     = (const float*)d_in[34];
  const float* Wp1     = (const float*)d_in[35];
  const float* bp1     = (const float*)d_in[36];
  const float* Wp2     = (const float*)d_in[37];
  const float* bp2     = (const float*)d_in[38];

  // ---- outputs: (preds, tmdo, fused) flat in return order ----
  float* preds_out = (float*)d_out;                       // B*T
  float* tmdo_out  = preds_out + BB * TT;                 // B*T*F
  float* fused_out = tmdo_out + BB * TT * FF;             // B*T*H

  // ---- workspace carve-out ----
  float* p = (float*)d_ws;
  auto alloc = [&](size_t n) { float* r = p; p += n; return r; };
  float* wsoft = alloc(FF * FF);
  float* memw  = alloc((size_t)BB * TT * FF);
  float* X1    = alloc((size_t)BB * TT * 3 * FF);
  float* hid1  = alloc((size_t)BB * TT * HH);
  float* Wkv   = alloc((size_t)HH * 2 * HH);
  float* bkv   = alloc(2 * HH);
  float* h     = alloc(BB * HH);
  float* c     = alloc(BB * HH);
  float* G     = alloc(BB * EE * EDD);
  float* rs    = alloc(BB * EE);
  float* h1e   = alloc(BB * EE * HH);
  float* impb  = alloc(BB * EE * HH);
  float* kkvv  = alloc(BB * EE * 2 * HH);
  float* qb    = alloc(BB * HH);
  float* evr   = alloc(BB * HH);
  float* ev    = alloc(BB * HH);
  float* gates = alloc(BB * 4 * HH);
  float* ind   = alloc(BB * HH);
  float* h2a   = alloc(BB * HH);
  float* h2b   = alloc(BB * HH);
  float* g1    = alloc(BB * HH);
  float* g2    = alloc(BB * HH);
  float* ph    = alloc(BB * (HH / 2));

  // ---- time-parallel precompute ----
  softmax_w_kernel<<<1, FF, 0, stream>>>(W_mcao, wsoft);
  build_wkv_kernel<<<(HH * 2 * HH + 255) / 256, 256, 0, stream>>>(Wk, bk, Wv, bv, Wkv, bkv);
  zero_kernel<<<(2 * BB * HH + 255) / 256, 256, 0, stream>>>(h, 2 * BB * HH); // h,c contiguous
  tmdo_kernel<<<BB * TT, FF, 0, stream>>>(x, coeffs, wsoft, Wg, bg, kappa,
                                          tmdo_out, memw, X1);
  // fused = relu(X1 @ Wf1 + bf1) @ Wf2 + bf2
  gemm(stream, X1, 3 * FF, Wf1, HH, 0, hid1, HH, bf1, nullptr, BB * TT, HH, 3 * FF, 0, 1);
  gemm(stream, hid1, HH, Wf2, HH, 0, fused_out, HH, bf2, nullptr, BB * TT, HH, HH, 0, 0);

  // ---- sequential scan over T ----
  for (int t = 0; t < TT; ++t) {
    gather_emb_kernel<<<BB * EE, 128, 0, stream>>>(events, td, E_emb, t, G, rs);
    gemm(stream, G, EDD, We1, HH, 0, h1e, HH, be1, nullptr, BB * EE, HH, EDD, 0, 1);
    gemm(stream, h1e, HH, We2, HH, 0, impb, HH, be2, rs, BB * EE, HH, HH, 0, 0);
    gemm(stream, impb, HH, Wkv, 2 * HH, 0, kkvv, 2 * HH, bkv, nullptr, BB * EE, 2 * HH, HH, 0, 0);
    gemm(stream, h, HH, Wq, HH, 0, qb, HH, bq, nullptr, BB, HH, HH, 0, 0);
    attn_kernel<<<BB * 4, 128, 0, stream>>>(qb, kkvv, evr);
    gemm(stream, evr, HH, Wo, HH, 0, ev, HH, bo, nullptr, BB, HH, HH, 0, 0);
    gemm(stream, fused_out + (size_t)t * HH, TT * HH, W_ih, HH, 1, gates, 4 * HH,
         b_ih, nullptr, BB, 4 * HH, HH, 0, 0);
    gemm(stream, h, HH, W_hh, HH, 1, gates, 4 * HH, b_hh, nullptr, BB, 4 * HH, HH, 1, 0);
    gemm(stream, x + (size_t)t * FF, TT * FF, Wip, HH, 0, ind, HH, bip, nullptr, BB, HH, FF, 0, 1);
    lstm_cell_kernel<<<(BB * HH) / 256, 256, 0, stream>>>(gates, c, h2a);
    gemm(stream, h2a, HH, Wig, HH, 0, g1, HH, big, nullptr, BB, HH, HH, 0, 0);
    gemm(stream, ind, HH, Wig + (size_t)HH * HH, HH, 0, g1, HH, nullptr, nullptr, BB, HH, HH, 1, 0);
    gate_mix_kernel<<<(BB * HH) / 256, 256, 0, stream>>>(h2a, ind, g1, h2b);
    gemm(stream, h2b, HH, Weg, HH, 0, g2, HH, beg, nullptr, BB, HH, HH, 0, 0);
    gemm(stream, ev, HH, Weg + (size_t)HH * HH, HH, 0, g2, HH, nullptr, nullptr, BB, HH, HH, 1, 0);
    gate_mix_kernel<<<(BB * HH) / 256, 256, 0, stream>>>(h2b, ev, g2, h);   // h = new state
    gemm(stream, h, HH, Wp1, HH / 2, 0, ph, HH / 2, bp1, nullptr, BB, HH / 2, HH, 0, 1);
    pred_kernel<<<BB, 256, 0, stream>>>(ph, Wp2, bp2, preds_out, t);
  }
}